// Projector_19954418057848
// MI455X (gfx1250) — compile-verified
//
#include <hip/hip_runtime.h>
#include <stdint.h>

// -----------------------------------------------------------------------------
// Point-splat z-buffer rasterizer for MI455X (gfx1250).
//
// Depth test via 64-bit packed atomic-min:
//   key = (float_bits(z) << 32) | point_index
// For z > 0 the IEEE-754 bit pattern is monotonically increasing with z, so
// min(key) selects the minimum-z point, and among equal z the minimum index —
// exactly the reference's stable argsort + min-rank semantics, and fully
// order-independent (deterministic under graph replay).
// -----------------------------------------------------------------------------

static constexpr uint64_t Z_SENTINEL = ~0ULL;

__global__ void init_depth_kernel(uint64_t* __restrict__ zbuf, int npix) {
    int i = blockIdx.x * blockDim.x + threadIdx.x;
    if (i < npix) zbuf[i] = Z_SENTINEL;
}

__global__ void project_kernel(const float* __restrict__ pts,
                               const float* __restrict__ Km,
                               uint64_t* __restrict__ zbuf,
                               int n, int width, int height) {
    // Camera intrinsics: broadcast loads, L0/L2 hit for every wave.
    const float fx = Km[0];
    const float cx = Km[2];
    const float fy = Km[4];
    const float cy = Km[5];

    const int stride = gridDim.x * blockDim.x;
    for (int i = blockIdx.x * blockDim.x + threadIdx.x; i < n; i += stride) {
        // Software prefetch one grid-stride ahead (global_prefetch_b8).
        int inext = i + stride;
        if (inext < n) __builtin_prefetch(&pts[3 * (size_t)inext], 0, 0);

        // Streamed once: non-temporal loads keep the 50 MB point array from
        // evicting the L2-resident depth buffer / colors.
        float x = __builtin_nontemporal_load(&pts[3 * (size_t)i + 0]);
        float y = __builtin_nontemporal_load(&pts[3 * (size_t)i + 1]);
        float z = __builtin_nontemporal_load(&pts[3 * (size_t)i + 2]);

        // round-to-nearest-even, matching jnp.round().astype(int32).
        // v_cvt_i32_f32 clamps overflow, so huge |t| safely fails bounds check.
        int u = __float2int_rn(fx * x / z + cx);
        int v = __float2int_rn(fy * y / z + cy);

        if (z > 0.0f && u >= 0 && u < width && v >= 0 && v < height) {
            uint64_t key = ((uint64_t)__float_as_uint(z) << 32) | (uint32_t)i;
            // Non-returning 64-bit atomic min -> global_atomic_min_u64
            // (STOREcnt path, no VGPR writeback).
            atomicMin((unsigned long long*)(zbuf + (size_t)v * width + u),
                      (unsigned long long)key);
        }
    }
}

__global__ void resolve_kernel(const uint64_t* __restrict__ zbuf,
                               const float* __restrict__ colors,
                               float* __restrict__ out, int npix) {
    int p = blockIdx.x * blockDim.x + threadIdx.x;
    if (p >= npix) return;

    uint64_t key = zbuf[p];
    float r = 0.0f, g = 0.0f, b = 0.0f;
    if (key != Z_SENTINEL) {
        int idx = (int)(uint32_t)(key & 0xFFFFFFFFu);
        r = colors[3 * (size_t)idx + 0];
        g = colors[3 * (size_t)idx + 1];
        b = colors[3 * (size_t)idx + 2];
    }
    // Written exactly once per replay; also clears the poisoned d_out for
    // pixels with no point. Non-temporal: output is never re-read.
    __builtin_nontemporal_store(r, &out[3 * (size_t)p + 0]);
    __builtin_nontemporal_store(g, &out[3 * (size_t)p + 1]);
    __builtin_nontemporal_store(b, &out[3 * (size_t)p + 2]);
}

extern "C" void kernel_launch(void* const* d_in, const int* in_sizes, int n_in,
                              void* d_out, int out_size, void* d_ws, size_t ws_size,
                              hipStream_t stream) {
    const float* pts    = (const float*)d_in[0];  // (N,3) f32
    const float* colors = (const float*)d_in[1];  // (N,3) f32
    const float* Km     = (const float*)d_in[2];  // (3,3) f32

    const int n      = in_sizes[0] / 3;   // 4194304
    const int npix   = out_size / 3;      // H*W = 1310720
    const int width  = 1280;              // compile-time constants in reference
    const int height = npix / width;      // 1024

    uint64_t* zbuf = (uint64_t*)d_ws;     // npix * 8 B = 10.5 MB scratch

    const int tb = 256;  // 8 wave32 waves per block

    // 1) Re-initialize depth keys every call (determinism under graph replay).
    init_depth_kernel<<<(npix + tb - 1) / tb, tb, 0, stream>>>(zbuf, npix);

    // 2) Project + packed 64-bit atomic-min depth test.
    //    Grid-stride with ~128K threads -> ~32 points/thread, prefetch ahead.
    int proj_blocks = 512;                // 512*256 = 131072 threads
    project_kernel<<<proj_blocks, tb, 0, stream>>>(pts, Km, zbuf, n, width, height);

    // 3) Resolve winners to RGB (writes every output pixel, incl. zeros).
    resolve_kernel<<<(npix + tb - 1) / tb, tb, 0, stream>>>(zbuf, colors,
                                                            (float*)d_out, npix);
}